// NEATLayer_2508260901061
// MI455X (gfx1250) — compile-verified
//
#include <hip/hip_runtime.h>

typedef __attribute__((ext_vector_type(2))) float v2f;
typedef __attribute__((ext_vector_type(4))) float v4f;
typedef __attribute__((ext_vector_type(8))) float v8f;

#define D_IN    64
#define N_EVAL  68
#define N_TOTAL 132
#define NPAD    80    // eval nodes padded to 5 groups of 16
#define SWS     148   // padded LDS row stride for W (>= 64+80 so triangular reads stay in-bounds, zero-filled)
#define SVS     84    // LDS row stride for per-wave node values (bank-conflict friendly)
#define WAVES_PER_BLOCK 2

// sigmoid(clip(5*(b + r*p), -60, 60)) = 1 / (1 + exp2(clip_u(FOLD*(b + r*p))))
// with FOLD = -5*log2(e); clamp in exp2 domain at +/- 60*log2(e).
#define FOLD   (-7.2134752044448169f)
#define UCLAMP (86.561702453337803f)

__device__ __forceinline__ float neat_sigmoid(float bF, float rF, float t) {
    float u = __builtin_fmaf(rF, t, bF);               // FOLD*(b + r*t)
    u = fminf(fmaxf(u, -UCLAMP), UCLAMP);
    float e = __builtin_amdgcn_exp2f(u);               // v_exp_f32 (TRANS)
    return __builtin_amdgcn_rcpf(1.0f + e);            // v_rcp_f32 (TRANS)
}

__global__ __launch_bounds__(64, 1)
void neat_wmma_kernel(const float* __restrict__ x,
                      const float* __restrict__ W,
                      const float* __restrict__ bias,
                      const float* __restrict__ resp,
                      float* __restrict__ out)
{
    __shared__ float sW[NPAD * SWS];
    __shared__ float sB[NPAD];   // holds FOLD*bias
    __shared__ float sR[NPAD];   // holds FOLD*response
    __shared__ float sV[WAVES_PER_BLOCK][16 * SVS];

    const int tid  = threadIdx.x;
    const int wave = tid >> 5;
    const int lane = tid & 31;

    // Stage W (topology-masked) into padded LDS; pad rows/cols are zero so
    // the padded nodes contribute nothing and triangular reads are safe.
    for (int idx = tid; idx < NPAD * SWS; idx += 64) {
        int r = idx / SWS, c = idx - r * SWS;
        sW[idx] = (r < N_EVAL && c < N_TOTAL) ? W[r * N_TOTAL + c] : 0.0f;
    }
    for (int idx = tid; idx < NPAD; idx += 64) {
        sB[idx] = (idx < N_EVAL) ? FOLD * bias[idx] : 0.0f;
        sR[idx] = (idx < N_EVAL) ? FOLD * resp[idx] : FOLD;
    }
    __syncthreads();

    const int  mrow  = lane & 15;   // A-fragment row / C-fragment N index
    const int  khalf = lane >> 4;   // which K pair (A) / which sample half (C)
    const long tile  = (long)blockIdx.x * WAVES_PER_BLOCK + wave;
    const long s0    = tile * 16;   // 16 samples per wave

    // Preload the 16x64 x-tile in WMMA A-fragment layout:
    // lane m holds row m, VGPR pair k holds K = 4k + 2*khalf + {0,1}.
    v2f xa[16];
    const float* xrowp = x + (s0 + mrow) * D_IN + khalf * 2;
    #pragma unroll
    for (int k = 0; k < 16; ++k)
        xa[k] = *(const v2f*)(xrowp + 4 * k);

    float* myV = &sV[wave][0];

    for (int g = 0; g < 5; ++g) {
        const int rowbase = (g * 16 + mrow) * SWS;
        v8f acc = {0.f, 0.f, 0.f, 0.f, 0.f, 0.f, 0.f, 0.f};

        // PRE[:, g-block] += X (16x64) * Wx[g-block]^T (64x16)  -- 16 K-steps
        const float* wb = &sW[rowbase + khalf * 2];
        #pragma unroll
        for (int k = 0; k < 16; ++k) {
            v2f b = *(const v2f*)(wb + 4 * k);
            acc = __builtin_amdgcn_wmma_f32_16x16x4_f32(false, xa[k], false, b,
                                                        (short)0, acc, false, false);
        }
        // contributions from earlier hidden groups: V_prev (16x16) * Whh^T
        for (int p = 0; p < g; ++p) {
            const float* va  = myV + mrow * SVS + p * 16 + khalf * 2;
            const float* wbh = &sW[rowbase + D_IN + p * 16 + khalf * 2];
            #pragma unroll
            for (int k = 0; k < 4; ++k) {
                v2f a = *(const v2f*)(va + 4 * k);
                v2f b = *(const v2f*)(wbh + 4 * k);
                acc = __builtin_amdgcn_wmma_f32_16x16x4_f32(false, a, false, b,
                                                            (short)0, acc, false, false);
            }
        }

        // In-group triangular weights: this lane's destination node is g*16+mrow.
        // W masking guarantees wl[i] == 0 for i >= mrow, so no lane predication.
        float wl[16];
        const float* wtr = &sW[rowbase + D_IN + g * 16];
        #pragma unroll
        for (int i = 0; i < 16; i += 4)
            *(v4f*)(wl + i) = *(const v4f*)(wtr + i);

        const float bl = sB[g * 16 + mrow];
        const float rl = sR[g * 16 + mrow];
        const int srcBase = (lane & 16) << 2;   // stay within this sample half

        // Sequential triangular sweep: broadcast node i's pre-activation from
        // lane i (per half-wave), sigmoid it, FMA into later nodes' lanes.
        #pragma unroll
        for (int i = 0; i < 16; ++i) {
            const float bi = sB[g * 16 + i];
            const float ri = sR[g * 16 + i];
            const int srcb = srcBase + (i << 2);
            #pragma unroll
            for (int r = 0; r < 8; ++r) {
                float t = __int_as_float(
                    __builtin_amdgcn_ds_bpermute(srcb, __float_as_int(acc[r])));
                float v = neat_sigmoid(bi, ri, t);
                acc[r] += wl[i] * v;
            }
        }

        // Finalize this group's node values; stash in LDS (A-fragment friendly)
        // for later groups' WMMA and for the output read-back.
        #pragma unroll
        for (int r = 0; r < 8; ++r) {
            float v = neat_sigmoid(bl, rl, acc[r]);
            myV[(r + khalf * 8) * SVS + g * 16 + mrow] = v;
        }
    }

    // Outputs are eval nodes 64..67 (group 4, locals 0..3): float4 per sample.
    if (lane < 16) {
        v4f o = *(const v4f*)(myV + lane * SVS + D_IN);
        *(v4f*)(out + (s0 + lane) * 4) = o;
    }
}

extern "C" void kernel_launch(void* const* d_in, const int* in_sizes, int n_in,
                              void* d_out, int out_size, void* d_ws, size_t ws_size,
                              hipStream_t stream) {
    const float* x    = (const float*)d_in[0];
    const float* W    = (const float*)d_in[1];
    const float* bias = (const float*)d_in[2];
    const float* resp = (const float*)d_in[3];
    float* out = (float*)d_out;

    const int B = in_sizes[0] / D_IN;                 // 262144
    const int blocks = B / (16 * WAVES_PER_BLOCK);    // 8192 blocks of 2 waves
    neat_wmma_kernel<<<blocks, 64, 0, stream>>>(x, W, bias, resp, out);
}